// MPN_65085934403703
// MI455X (gfx1250) — compile-verified
//
#include <hip/hip_runtime.h>
#include <stdint.h>

typedef float v2f __attribute__((ext_vector_type(2)));
typedef float v8f __attribute__((ext_vector_type(8)));

__device__ __forceinline__ v8f wmma4(v2f a, v2f b, v8f c) {
  // V_WMMA_F32_16X16X4_F32: D = A(16x4) * B(4x16) + C(16x16), fp32 throughout.
  return __builtin_amdgcn_wmma_f32_16x16x4_f32(false, a, false, b, (short)0, c,
                                               false, false);
}

// CDNA5 async global->LDS copy, 16B per lane. Tracked by ASYNCcnt.
__device__ __forceinline__ void async_copy_b128(uint32_t lds_byte_off,
                                                const void* gaddr) {
  asm volatile("global_load_async_to_lds_b128 %0, %1, off"
               :: "v"(lds_byte_off), "v"(gaddr) : "memory");
}
__device__ __forceinline__ void async_wait0() {
  asm volatile("s_wait_asynccnt 0x0" ::: "memory");
}

// Undirected edge expansion: g in [0,2E). First half: (row,col)=(ei0,ei1);
// second half reversed with edge_attr cols 0,2 sign-flipped (handled by caller).
__device__ __forceinline__ void edge_rc(const int* __restrict__ ei, int E, int g,
                                        int& r, int& c, int& eaI, bool& flip) {
  if (g < E) { r = ei[g];       c = ei[E + g]; eaI = g;     flip = false; }
  else       { int t = g - E; r = ei[E + t]; c = ei[t]; eaI = t; flip = true; }
}

__global__ void __launch_bounds__(256) k_zero(float* __restrict__ p, int n) {
  int i = blockIdx.x * blockDim.x + threadIdx.x;
  if (i < n) p[i] = 0.f;
}

__global__ void __launch_bounds__(256) k_degree(const int* __restrict__ ei,
                                                float* __restrict__ deg, int E) {
  int g = blockIdx.x * blockDim.x + threadIdx.x;
  if (g >= 2 * E) return;
  int r, c, eaI; bool f;
  edge_rc(ei, E, g, r, c, eaI, f);
  atomicAdd(&deg[c], 1.0f);
}

__global__ void __launch_bounds__(256) k_dinv(float* __restrict__ deg, int n) {
  int i = blockIdx.x * blockDim.x + threadIdx.x;
  if (i >= n) return;
  float d = deg[i];
  deg[i] = (d > 0.f) ? rsqrtf(fmaxf(d, 1.f)) : 0.f;
}

__global__ void __launch_bounds__(256) k_norm(const int* __restrict__ ei,
                                              const float* __restrict__ dinv,
                                              float* __restrict__ nrm, int E) {
  int g = blockIdx.x * blockDim.x + threadIdx.x;
  if (g >= 2 * E) return;
  int r, c, eaI; bool f;
  edge_rc(ei, E, g, r, c, eaI, f);
  nrm[g] = dinv[r] * dinv[c];
}

// One normalized sparse hop: dst[col] += src[row] * norm. 8 threads/edge, float4 each.
__global__ void __launch_bounds__(256) k_prop(const float* __restrict__ src,
                                              float* __restrict__ dst,
                                              const int* __restrict__ ei,
                                              const float* __restrict__ nrm, int E) {
  int idx = blockIdx.x * blockDim.x + threadIdx.x;
  int total = 2 * E * 8;
  if (idx >= total) return;
  int e = idx >> 3;
  int q = (idx & 7) * 4;
  int r, c, eaI; bool f;
  edge_rc(ei, E, e, r, c, eaI, f);
  float nm = nrm[e];
  const float4 v = *(const float4*)(src + r * 32 + q);
  float* d = dst + c * 32 + q;
  atomicAdd(d + 0, v.x * nm);
  atomicAdd(d + 1, v.y * nm);
  atomicAdd(d + 2, v.z * nm);
  atomicAdd(d + 3, v.w * nm);
}

// Edge MLP: relu([x[col],x[row],ea] @ W1 + b1) @ W2 + b2, scatter-added at col.
// One wave handles 16 edges; M=16 (edges), K=40 then 32, N=32 (two 16-wide tiles).
// Weights are async-copied to LDS then re-laid-out into WMMA B-fragment order:
// Wf[kb][n][kk] so each lane's fragment is one 8B ds_load_b64.
__global__ void __launch_bounds__(256) k_edge_mlp(
    const float* __restrict__ x, const int* __restrict__ ei,
    const float* __restrict__ eattr,
    const float* __restrict__ w1, const float* __restrict__ b1g,
    const float* __restrict__ w2, const float* __restrict__ b2g,
    float* __restrict__ h, int E) {
  __shared__ __align__(16) float sStage[2304];   // raw w1 (1280) + w2 (1024)
  __shared__ __align__(16) float sW1f[1280];     // fragment layout, 10 K-blocks
  __shared__ __align__(16) float sW2f[1024];     // fragment layout, 8 K-blocks
  __shared__ float sB1[32];
  __shared__ float sB2[32];
  __shared__ float sZ[8][16 * 42];  // stride 42: conflict-free, 8B aligned pairs
  __shared__ float sT[8][16 * 34];  // stride 34: conflict-free, 8B aligned pairs
  __shared__ int   sRow[8][16];
  __shared__ int   sCol[8][16];

  // Async global->LDS stream of both weight matrices (16B granules).
  {
    uint32_t base = (uint32_t)(uintptr_t)&sStage[0];
    for (int i = threadIdx.x * 4; i < 1280; i += 1024)
      async_copy_b128(base + (uint32_t)i * 4u, w1 + i);
    for (int i = threadIdx.x * 4; i < 1024; i += 1024)
      async_copy_b128(base + (1280u + (uint32_t)i) * 4u, w2 + i);
    async_wait0();
  }
  if (threadIdx.x < 32) { sB1[threadIdx.x] = b1g[threadIdx.x];
                          sB2[threadIdx.x] = b2g[threadIdx.x]; }
  __syncthreads();

  // Re-layout [K,N] -> fragment order [kb][n][kk] (one-time per block).
  for (int i = threadIdx.x; i < 1280; i += 256) {
    int kb = i >> 7, j = i & 127, n = j >> 2, kk = j & 3;
    sW1f[i] = sStage[(4 * kb + kk) * 32 + n];
  }
  for (int i = threadIdx.x; i < 1024; i += 256) {
    int kb = i >> 7, j = i & 127, n = j >> 2, kk = j & 3;
    sW2f[i] = sStage[1280 + (4 * kb + kk) * 32 + n];
  }
  __syncthreads();

  const int wid  = threadIdx.x >> 5;
  const int lane = threadIdx.x & 31;
  const int E2   = 2 * E;
  const int ge0  = (blockIdx.x * 8 + wid) * 16;
  if (ge0 >= E2) return;

  if (lane < 16) {
    int g = ge0 + lane; if (g >= E2) g = E2 - 1;
    int r, c, eaI; bool f;
    edge_rc(ei, E, g, r, c, eaI, f);
    sRow[wid][lane] = r; sCol[wid][lane] = c;
  }
  __builtin_amdgcn_wave_barrier();

  // Gather 16 edges x 40 features into LDS (640 slots, 20 per lane).
  for (int s = lane; s < 640; s += 32) {
    int el = s / 40;
    int f  = s - el * 40;
    int g  = ge0 + el; if (g >= E2) g = E2 - 1;
    bool flip = g >= E;
    int eaI = flip ? (g - E) : g;
    int r = sRow[wid][el], c = sCol[wid][el];
    float v;
    if (f < 16)      v = x[c * 16 + f];
    else if (f < 32) v = x[r * 16 + (f - 16)];
    else {
      int fe = f - 32;
      v = eattr[eaI * 8 + fe];
      if (flip && (fe == 0 || fe == 2)) v = -v;
    }
    sZ[wid][el * 42 + f] = v;
  }
  __builtin_amdgcn_wave_barrier();

  const int ml = lane & 15;          // A: M row / B,C,D: N column within tile
  const int kh = (lane >> 4) << 1;   // lanes 16-31 carry K=k0+2,k0+3

  // Layer 1: K = 40 -> 10 K-blocks, two N tiles.
  v8f a0 = {}, a1 = {};
  const float* zrow = &sZ[wid][ml * 42];
#pragma unroll
  for (int kb = 0; kb < 10; ++kb) {
    v2f a  = *(const v2f*)(zrow + 4 * kb + kh);
    v2f bA = *(const v2f*)(sW1f + kb * 128 + ml * 4 + kh);
    v2f bB = *(const v2f*)(sW1f + kb * 128 + (16 + ml) * 4 + kh);
    a0 = wmma4(a, bA, a0);
    a1 = wmma4(a, bB, a1);
  }

  // bias + relu, transpose D->A layout through LDS.
#pragma unroll
  for (int i = 0; i < 8; ++i) {
    int m = (lane < 16) ? i : (8 + i);
    sT[wid][m * 34 + ml]      = fmaxf(a0[i] + sB1[ml],      0.f);
    sT[wid][m * 34 + 16 + ml] = fmaxf(a1[i] + sB1[16 + ml], 0.f);
  }
  __builtin_amdgcn_wave_barrier();

  // Layer 2: K = 32 -> 8 K-blocks.
  v8f d0 = {}, d1 = {};
  const float* trow = &sT[wid][ml * 34];
#pragma unroll
  for (int kb = 0; kb < 8; ++kb) {
    v2f a  = *(const v2f*)(trow + 4 * kb + kh);
    v2f bA = *(const v2f*)(sW2f + kb * 128 + ml * 4 + kh);
    v2f bB = *(const v2f*)(sW2f + kb * 128 + (16 + ml) * 4 + kh);
    d0 = wmma4(a, bA, d0);
    d1 = wmma4(a, bB, d1);
  }

  // Scatter-add message rows into h[col].
#pragma unroll
  for (int i = 0; i < 8; ++i) {
    int m = (lane < 16) ? i : (8 + i);
    if (ge0 + m >= E2) continue;
    int c = sCol[wid][m];
    atomicAdd(&h[c * 32 + ml],      d0[i] + sB2[ml]);
    atomicAdd(&h[c * 32 + 16 + ml], d1[i] + sB2[16 + ml]);
  }
}

// Fused TAGConv combine: out = relu?( b + sum_s Z_s @ W[s] ). One wave = 16 nodes.
__global__ void __launch_bounds__(256) k_tag(
    const float* __restrict__ z0, const float* __restrict__ z1,
    const float* __restrict__ z2, const float* __restrict__ z3,
    const float* __restrict__ w, const float* __restrict__ bias,
    float* __restrict__ out, int n, int relu) {
  __shared__ __align__(16) float sStage[4096];
  __shared__ __align__(16) float swf[4096];   // fragment layout, 4 x 8 K-blocks
  __shared__ float sb[32];

  {
    uint32_t base = (uint32_t)(uintptr_t)&sStage[0];
    for (int i = threadIdx.x * 4; i < 4096; i += 1024)
      async_copy_b128(base + (uint32_t)i * 4u, w + i);
    async_wait0();
  }
  if (threadIdx.x < 32) sb[threadIdx.x] = bias[threadIdx.x];
  __syncthreads();

  for (int i = threadIdx.x; i < 4096; i += 256) {
    int s = i >> 10, rem = i & 1023;
    int kb = rem >> 7, j = rem & 127, nn = j >> 2, kk = j & 3;
    swf[i] = sStage[s * 1024 + (4 * kb + kk) * 32 + nn];
  }
  __syncthreads();

  const int wid  = threadIdx.x >> 5;
  const int lane = threadIdx.x & 31;
  const int tile = blockIdx.x * 8 + wid;
  const int tiles = (n + 15) >> 4;
  if (tile >= tiles) return;
  const int base = tile * 16;
  const int ml = lane & 15;
  const int kh = (lane >> 4) << 1;
  int mrow = base + ml; if (mrow >= n) mrow = n - 1;  // clamp keeps EXEC full

  const float* zs[4] = {z0, z1, z2, z3};
  v8f acc0 = {}, acc1 = {};
#pragma unroll
  for (int s = 0; s < 4; ++s) {
    const float* Z = zs[s];
    const float* W = swf + s * 1024;
#pragma unroll
    for (int kb = 0; kb < 8; ++kb) {
      v2f a  = *(const v2f*)(Z + mrow * 32 + 4 * kb + kh);
      v2f bA = *(const v2f*)(W + kb * 128 + ml * 4 + kh);
      v2f bB = *(const v2f*)(W + kb * 128 + (16 + ml) * 4 + kh);
      acc0 = wmma4(a, bA, acc0);
      acc1 = wmma4(a, bB, acc1);
    }
  }
#pragma unroll
  for (int i = 0; i < 8; ++i) {
    int m = base + ((lane < 16) ? i : (8 + i));
    if (m >= n) continue;
    float v0 = acc0[i] + sb[ml];
    float v1 = acc1[i] + sb[16 + ml];
    if (relu) { v0 = fmaxf(v0, 0.f); v1 = fmaxf(v1, 0.f); }
    out[m * 32 + ml]      = v0;
    out[m * 32 + 16 + ml] = v1;
  }
}

// Final layer: [N,32] x [32,4] x 4 sources + bias -> [N,4]. Scalar is appropriate.
__global__ void __launch_bounds__(256) k_final(
    const float* __restrict__ z0, const float* __restrict__ z1,
    const float* __restrict__ z2, const float* __restrict__ z3,
    const float* __restrict__ w, const float* __restrict__ bias,
    float* __restrict__ out, int n) {
  __shared__ float sw[4 * 32 * 4];
  __shared__ float sb[4];
  for (int i = threadIdx.x; i < 512; i += 256) sw[i] = w[i];
  if (threadIdx.x < 4) sb[threadIdx.x] = bias[threadIdx.x];
  __syncthreads();
  int node = blockIdx.x * blockDim.x + threadIdx.x;
  if (node >= n) return;
  float a0 = sb[0], a1 = sb[1], a2 = sb[2], a3 = sb[3];
  const float* zs[4] = {z0, z1, z2, z3};
#pragma unroll
  for (int s = 0; s < 4; ++s) {
    const float* Z = zs[s] + node * 32;
    const float* W = sw + s * 128;
#pragma unroll
    for (int c = 0; c < 32; ++c) {
      float zv = Z[c];
      a0 += zv * W[c * 4 + 0];
      a1 += zv * W[c * 4 + 1];
      a2 += zv * W[c * 4 + 2];
      a3 += zv * W[c * 4 + 3];
    }
  }
  float* o = out + node * 4;
  o[0] = a0; o[1] = a1; o[2] = a2; o[3] = a3;
}

extern "C" void kernel_launch(void* const* d_in, const int* in_sizes, int n_in,
                              void* d_out, int out_size, void* d_ws, size_t ws_size,
                              hipStream_t stream) {
  const float* x   = (const float*)d_in[0];
  const int*   ei  = (const int*)  d_in[1];
  const float* ea  = (const float*)d_in[2];
  const float* w1  = (const float*)d_in[3];
  const float* b1  = (const float*)d_in[4];
  const float* w2  = (const float*)d_in[5];
  const float* b2  = (const float*)d_in[6];
  const float* c0w = (const float*)d_in[7];
  const float* c0b = (const float*)d_in[8];
  const float* c1w = (const float*)d_in[9];
  const float* c1b = (const float*)d_in[10];
  const float* c2w = (const float*)d_in[11];
  const float* c2b = (const float*)d_in[12];

  const int N  = in_sizes[0] / 16;   // 50000
  const int E  = in_sizes[1] / 2;    // 800000 directed; doubled by undirect
  const int E2 = 2 * E;

  float* ws = (float*)d_ws;
  size_t off = 0;
  float* deg = ws + off; off += (size_t)((N + 63) / 64) * 64;  // becomes dinv in place
  float* nrm = ws + off; off += (size_t)E2;
  float* h   = ws + off; off += (size_t)N * 32;
  float* t1  = ws + off; off += (size_t)N * 32;
  float* t2  = ws + off; off += (size_t)N * 32;
  float* t3  = ws + off; off += (size_t)N * 32;
  float* h2  = ws + off; off += (size_t)N * 32;

  const int B = 256;
  auto cdiv = [](int a, int b) { return (a + b - 1) / b; };

  // Degree -> dinv -> per-edge norm.
  k_zero  <<<cdiv(N, B), B, 0, stream>>>(deg, N);
  k_degree<<<cdiv(E2, B), B, 0, stream>>>(ei, deg, E);
  k_dinv  <<<cdiv(N, B), B, 0, stream>>>(deg, N);
  k_norm  <<<cdiv(E2, B), B, 0, stream>>>(ei, deg, nrm, E);

  // Edge MLP + scatter into h.
  k_zero<<<cdiv(N * 32, B), B, 0, stream>>>(h, N * 32);
  int waves = cdiv(E2, 16);
  k_edge_mlp<<<cdiv(waves, 8), B, 0, stream>>>(x, ei, ea, w1, b1, w2, b2, h, E);

  const int propT = E2 * 8;
  const int tagBlocks = cdiv((N + 15) / 16, 8);

  auto hops = [&](const float* zin) {
    k_zero<<<cdiv(N * 32, B), B, 0, stream>>>(t1, N * 32);
    k_prop<<<cdiv(propT, B), B, 0, stream>>>(zin, t1, ei, nrm, E);
    k_zero<<<cdiv(N * 32, B), B, 0, stream>>>(t2, N * 32);
    k_prop<<<cdiv(propT, B), B, 0, stream>>>(t1, t2, ei, nrm, E);
    k_zero<<<cdiv(N * 32, B), B, 0, stream>>>(t3, N * 32);
    k_prop<<<cdiv(propT, B), B, 0, stream>>>(t2, t3, ei, nrm, E);
  };

  // TAGConv layer 0: h -> h2 (relu)
  hops(h);
  k_tag<<<tagBlocks, B, 0, stream>>>(h, t1, t2, t3, c0w, c0b, h2, N, 1);
  // TAGConv layer 1: h2 -> h (relu)
  hops(h2);
  k_tag<<<tagBlocks, B, 0, stream>>>(h2, t1, t2, t3, c1w, c1b, h, N, 1);
  // TAGConv layer 2: h -> out [N,4], no relu
  hops(h);
  k_final<<<cdiv(N, B), B, 0, stream>>>(h, t1, t2, t3, c2w, c2b, (float*)d_out, N);
}